// LocallyConnected2dV2_35072702939571
// MI455X (gfx1250) — compile-verified
//
#include <hip/hip_runtime.h>
#include <stdint.h>

// Locally-connected 2D (untied conv): out[i,j] = sum_{p,q} x[i+p,j+q]*W[i,j,p,q]
// Memory-bound by the one-shot 171MB W stream (AI ~0.5 FLOP/B) -> stream W with
// NT hints, stage reused x tile into LDS via gfx1250 async global->LDS copies.

#define IN_H 450
#define IN_W 450
#define KH   15
#define KW   15
#define OH   436
#define OW   436

#define TILE     16                 // output tile is TILE x TILE
#define STRIP    4                  // outputs per thread (1 row x 4 cols)
#define NTHREADS 64                 // (TILE rows) x (TILE/STRIP strips) = 16*4
#define XT_H     (TILE + KH - 1)    // 30
#define XT_W     (TILE + KW - 1)    // 30
#define XT_PITCH 32                 // padded row pitch (floats)

__global__ __launch_bounds__(NTHREADS)
void lc2d_stream_kernel(const float* __restrict__ x,
                        const float* __restrict__ W,
                        float* __restrict__ out)
{
    __shared__ float xt[XT_H * XT_PITCH];   // 3.75 KB

    const int tid = threadIdx.x;
    const int i0  = blockIdx.y * TILE;
    const int j0  = blockIdx.x * TILE;

    // ---- Stage the 30x30 input patch into LDS with async global->LDS copies
    // (gfx1250 GLOBAL_LOAD_ASYNC_TO_LDS_B32, tracked by ASYNCcnt). ----
    for (int idx = tid; idx < XT_H * XT_W; idx += NTHREADS) {
        const int rr = idx / XT_W;
        const int cc = idx - rr * XT_W;
        const int gi = i0 + rr;
        const int gj = j0 + cc;
        if (gi < IN_H && gj < IN_W) {
            const float* gptr   = x + (size_t)gi * IN_W + gj;
            unsigned     ldsoff = (unsigned)(uintptr_t)&xt[rr * XT_PITCH + cc];
            asm volatile("global_load_async_to_lds_b32 %0, %1, off"
                         :: "v"(ldsoff), "v"(gptr) : "memory");
        }
    }
    asm volatile("s_wait_asynccnt 0x0" ::: "memory");
    __syncthreads();

    // ---- Each thread computes a 1x4 strip of outputs. ----
    const int r   = tid >> 2;            // 0..15   output row within tile
    const int c0  = (tid & 3) * STRIP;   // 0,4,8,12 first output col within tile
    const int oi  = i0 + r;
    const int ojb = j0 + c0;

    // Clamp W bases for out-of-range strips so streaming stays in-bounds
    // (results for clamped lanes are discarded at the store).
    const float* wp[STRIP];
#pragma unroll
    for (int c = 0; c < STRIP; ++c) {
        const int ii = (oi      < OH) ? oi      : (OH - 1);
        const int jj = (ojb + c < OW) ? ojb + c : (OW - 1);
        wp[c] = W + ((size_t)ii * OW + jj) * (KH * KW);
    }

    float acc[STRIP] = {0.f, 0.f, 0.f, 0.f};

    for (int p = 0; p < KH; ++p) {
        // x row segment shared by the 4 outputs and all 15 taps.
        // Base (r+p)*32 + c0 is 16B-aligned -> vectorizable ds loads.
        float xv[STRIP + KW - 1];        // 18 floats
        const float* xrow = &xt[(r + p) * XT_PITCH + c0];
#pragma unroll
        for (int t = 0; t < STRIP + KW - 1; ++t) xv[t] = xrow[t];

#pragma unroll
        for (int q = 0; q < KW; ++q) {
#pragma unroll
            for (int c = 0; c < STRIP; ++c) {
                // W is single-use: non-temporal load (th:NT), don't pollute L2.
                const float w = __builtin_nontemporal_load(wp[c] + p * KW + q);
                acc[c] = fmaf(xv[q + c], w, acc[c]);
            }
        }
    }

    if (oi < OH) {
#pragma unroll
        for (int c = 0; c < STRIP; ++c) {
            if (ojb + c < OW) {
                __builtin_nontemporal_store(acc[c], out + (size_t)oi * OW + (ojb + c));
            }
        }
    }
}

extern "C" void kernel_launch(void* const* d_in, const int* in_sizes, int n_in,
                              void* d_out, int out_size, void* d_ws, size_t ws_size,
                              hipStream_t stream) {
    (void)in_sizes; (void)n_in; (void)out_size; (void)d_ws; (void)ws_size;
    const float* x = (const float*)d_in[0];
    const float* W = (const float*)d_in[1];
    float*       o = (float*)d_out;

    dim3 grid((OW + TILE - 1) / TILE, (OH + TILE - 1) / TILE);  // 28 x 28
    lc2d_stream_kernel<<<grid, NTHREADS, 0, stream>>>(x, W, o);
}